// GeometricAwareHypAggAtt_29240137351634
// MI455X (gfx1250) — compile-verified
//
#include <hip/hip_runtime.h>
#include <hip/hip_bf16.h>
#include <math.h>

// CDNA5 / gfx1250, wave32.
typedef __attribute__((ext_vector_type(2))) float v2f;
typedef __attribute__((ext_vector_type(8))) float v8f;

#define TPB 128                 // 4 waves/block
#define WPB (TPB / 32)
#define ROWP 68                 // 16x64 tile padded to 68 floats/row: 16B-aligned rows, conflict-free col reads
#define TILE_F (16 * ROWP)      // 1088 floats
#define DIAGP 17
#define DIAG_F (16 * DIAGP)     // 272 floats
#define WAVE_LDS (2 * TILE_F + DIAG_F)  // 2448 floats per wave

// ---------------------------------------------------------------------------
// Pass 1: per-node squared norms into d_ws, and init rowsum output to 1e-10.
// ---------------------------------------------------------------------------
__global__ __launch_bounds__(256) void hypagg_norms_init(
    const float* __restrict__ x, float* __restrict__ norms,
    float* __restrict__ rowsum, int nN)
{
    int i = blockIdx.x * blockDim.x + threadIdx.x;
    if (i >= nN) return;
    const float4* p = (const float4*)(x + (size_t)i * 64);
    float s = 0.0f;
#pragma unroll
    for (int j = 0; j < 16; ++j) {
        float4 v = p[j];
        s += v.x * v.x + v.y * v.y + v.z * v.z + v.w * v.w;
    }
    norms[i]  = s;
    rowsum[i] = 1e-10f;   // reference adds 1e-10 to the segment sum
}

// ---------------------------------------------------------------------------
// Pass 2: per-edge scores. Each wave handles 16 edges per iteration.
//   - gather 16 src rows + 16 dst rows (float4 global loads) into LDS tiles
//   - 16 x V_WMMA_F32_16X16X4_F32 accumulate S(16x64) * D^T(64x16) in fp32
//   - diagonal (m,m) = dot(src_m, dst_m); finish scalar hyperbolic math
// A-frag layout (ISA 7.12.2, 32-bit A 16x4): lane<16 -> row=lane, K={0,1};
// lane>=16 -> row=lane-16, K={2,3}. B 4x16 mirrors it with B = D^T, so both
// fragments are identical float2 row reads from the LDS tiles.
// ---------------------------------------------------------------------------
__global__ __launch_bounds__(TPB) void hypagg_edge_kernel(
    const float* __restrict__ x,
    const int* __restrict__ e0, const int* __restrict__ e1,
    const float* __restrict__ norms,
    const float* __restrict__ beta, const float* __restrict__ con,
    const int* __restrict__ cptr,
    float* __restrict__ edge_out, float* __restrict__ rowsum,
    int nE)
{
    __shared__ float lds[WPB * WAVE_LDS];

    const int tid  = threadIdx.x;
    const int wave = tid >> 5;
    const int lane = tid & 31;

    float* S = lds + wave * WAVE_LDS;   // 16 x 64 src rows (padded)
    float* D = S + TILE_F;              // 16 x 64 dst rows (padded)
    float* G = D + TILE_F;              // 16 x 16 C tile for diag extraction

    const float cf   = (float)cptr[0];
    const float b0   = beta[0];
    const float c0   = con[0];
    const float sqc  = sqrtf(cf);
    const float dscl = 2.0f / sqc;

    const int batches = (nE + 15) >> 4;
    const int gWaves  = (gridDim.x * blockDim.x) >> 5;
    const int waveId  = (blockIdx.x * blockDim.x + tid) >> 5;

    for (int b = waveId; b < batches; b += gWaves) {
        const int eBase = b << 4;
        const int myE   = eBase + lane;

        int si = 0, di = 0;
        if (lane < 16) {
            int ei = (myE < nE) ? myE : (nE - 1);   // clamp tail (stores are guarded)
            si = e0[ei];
            di = e1[ei];
        }

        // Gather: 2 lanes per row, 128B (8 x float4) per lane per tile.
        const int r    = lane >> 1;
        const int half = lane & 1;
        const int rowS = __shfl(si, r, 32);
        const int rowD = __shfl(di, r, 32);
        const float4* ps = (const float4*)(x + (size_t)rowS * 64 + half * 32);
        const float4* pd = (const float4*)(x + (size_t)rowD * 64 + half * 32);
        float* sRow = S + r * ROWP + half * 32;
        float* dRow = D + r * ROWP + half * 32;
#pragma unroll
        for (int j = 0; j < 8; ++j) {
            float4 v = ps[j];
            float4 w = pd[j];
            *(float4*)(sRow + 4 * j) = v;
            *(float4*)(dRow + 4 * j) = w;
        }
        __syncthreads();   // order LDS tile writes vs cross-lane fragment reads

        // 16 chained fp32 WMMAs: K = 64 in steps of 4. EXEC is all-1s here.
        v8f acc = {};
        const int rr = lane & 15;
        const int kb = (lane >> 4) * 2;
#pragma unroll
        for (int k = 0; k < 16; ++k) {
            const int ko = 4 * k + kb;
            v2f a  = *(const v2f*)(S + rr * ROWP + ko);
            v2f bb = *(const v2f*)(D + rr * ROWP + ko);
            acc = __builtin_amdgcn_wmma_f32_16x16x4_f32(
                false, a, false, bb, (short)0, acc, false, false);
        }

        // Spill C tile to LDS; C layout: vgpr v, lane<16 -> (M=v, N=lane),
        // lane>=16 -> (M=v+8, N=lane-16). We need diag (m,m).
        const int M0 = (lane < 16) ? 0 : 8;
        const int N  = lane & 15;
#pragma unroll
        for (int v = 0; v < 8; ++v)
            G[(M0 + v) * DIAGP + N] = acc[v];
        __syncthreads();

        float xy = 0.0f;
        if (lane < 16) xy = G[lane * DIAGP + lane];

        if (lane < 16 && myE < nE) {
            const float n1 = norms[si];
            const float n2 = norms[di];
            // mobius_add(-p1, p2): num = -a1*p1 + b1*p2
            const float a1 = 1.0f - 2.0f * cf * xy + cf * n2;
            const float b1 = 1.0f - cf * n1;
            float num2 = a1 * a1 * n1 - 2.0f * a1 * b1 * xy + b1 * b1 * n2;
            float den  = fmaxf(1.0f - 2.0f * cf * xy + cf * cf * n1 * n2, 1e-15f);
            float nrm  = sqrtf(fmaxf(num2, 0.0f)) / den;
            float t    = fminf(sqc * nrm, 1.0f - 1e-7f);      // clip upper only
            float art  = 0.5f * logf((1.0f + t) / (1.0f - t)); // artanh
            float dd   = dscl * art;
            float ee   = tanhf(b0 * (dd * dd) + c0);
            edge_out[myE] = ee;
            atomicAdd(&rowsum[si], fabsf(ee));  // segment_sum over edge[0]
        }
        __syncthreads();   // protect tile reuse on next iteration
    }
}

// ---------------------------------------------------------------------------
extern "C" void kernel_launch(void* const* d_in, const int* in_sizes, int n_in,
                              void* d_out, int out_size, void* d_ws, size_t ws_size,
                              hipStream_t stream)
{
    const float* x    = (const float*)d_in[0];
    const int*   edge = (const int*)d_in[1];   // JAX default x64-off -> int32
    const float* beta = (const float*)d_in[2];
    const float* con  = (const float*)d_in[3];
    const int*   cp   = (const int*)d_in[4];

    const int nN = in_sizes[0] / 64;
    const int nE = in_sizes[1] / 2;

    float* out      = (float*)d_out;
    float* edge_out = out;           // [E]
    float* rowsum   = out + nE;      // [n]
    float* norms    = (float*)d_ws;  // n floats of scratch

    hypagg_norms_init<<<(nN + 255) / 256, 256, 0, stream>>>(x, norms, rowsum, nN);

    const int batches = (nE + 15) / 16;
    int blocks = (batches + WPB - 1) / WPB;
    if (blocks > 4096) blocks = 4096;   // grid-stride over 16-edge batches
    hypagg_edge_kernel<<<blocks, TPB, 0, stream>>>(
        x, edge, edge + nE, norms, beta, con, cp, edge_out, rowsum, nE);
}